// GraphConvolutionLayer_11914239279237
// MI455X (gfx1250) — compile-verified
//
#include <hip/hip_runtime.h>

#define F_DIM   128
#define BATCH   2
#define N_NODES 50000
#define AS_STRIDE 132   // 16x128 A tile padded to avoid LDS bank conflicts

typedef __attribute__((ext_vector_type(2))) float v2f;
typedef __attribute__((ext_vector_type(8))) float v8f;

// ---------------------------------------------------------------------------
// Kernel 1: zero the aggregation buffer (aliased onto d_out) and init degree=1
// ---------------------------------------------------------------------------
__global__ void gcn_init_kernel(float* __restrict__ agg,
                                float* __restrict__ degree,
                                long long total, int n_nodes) {
    long long idx    = (long long)blockIdx.x * blockDim.x + threadIdx.x;
    long long stride = (long long)gridDim.x * blockDim.x;
    for (long long i = idx; i < total; i += stride) agg[i] = 0.0f;
    for (long long i = idx; i < n_nodes; i += stride) degree[i] = 1.0f;
}

// ---------------------------------------------------------------------------
// Kernel 2: one wave32 per edge. 32 lanes x float4 = 128 features per batch.
// Atomic f32 adds land in L2 (agg is 51MB, fits the 192MB L2).
// ---------------------------------------------------------------------------
__global__ __launch_bounds__(256)
void gcn_scatter_kernel(const float* __restrict__ nodes,
                        const long long* __restrict__ edges,
                        float* __restrict__ agg,
                        float* __restrict__ degree,
                        int n_edges) {
    const int gid  = blockIdx.x * blockDim.x + threadIdx.x;
    const int edge = gid >> 5;
    const int lane = gid & 31;
    if (edge >= n_edges) return;

    const long long src = edges[2 * edge + 0];
    const long long dst = edges[2 * edge + 1];

    if (lane == 0) atomicAdd(&degree[dst], 1.0f);

#pragma unroll
    for (int b = 0; b < BATCH; ++b) {
        const long long sbase = ((long long)b * N_NODES + src) * F_DIM;
        const long long dbase = ((long long)b * N_NODES + dst) * F_DIM;
        const float4 v = ((const float4*)(nodes + sbase))[lane];
        float* dp = agg + dbase + lane * 4;
        atomicAdd(dp + 0, v.x);
        atomicAdd(dp + 1, v.y);
        atomicAdd(dp + 2, v.z);
        atomicAdd(dp + 3, v.w);
    }
}

// ---------------------------------------------------------------------------
// Kernel 3: per block: 16 rows x 128 cols of output.
//   - stage 16x128 A tile into LDS, scaled by 1/degree (normalization fused)
//   - 8 waves, each owns a 16-col tile; 32 x V_WMMA_F32_16X16X4_F32 over K=128
//   - B fragments read straight from global weight (64KB, L2/WGP$ resident)
//   - bias + ReLU fused into the store; output overwrites the same 16 rows
//     of the inout buffer (safe: staged via LDS + barrier before overwrite)
// ---------------------------------------------------------------------------
__global__ __launch_bounds__(256)
void gcn_gemm_kernel(float* __restrict__ inout,          // agg in, result out
                     const float* __restrict__ weight,   // [K=128][O=128]
                     const float* __restrict__ bias,     // [128]
                     const float* __restrict__ degree) { // [N]
    __shared__ float As[16 * AS_STRIDE];

    const int tid = threadIdx.x;
    const int m0  = blockIdx.x * 16;   // global row base in [0, B*N)

    __builtin_prefetch(weight, 0, 3);  // gfx1250 global_prefetch_b8

    // Load + normalize A tile: 16 rows x 128 f32 = 512 float4, 2 per thread.
    for (int i = tid; i < 16 * 32; i += 256) {
        const int row = i >> 5;        // 0..15
        const int c4  = i & 31;        // float4 column
        const int gm  = m0 + row;
        const int node = gm % N_NODES;
        const float inv = 1.0f / degree[node];
        const float4 v = ((const float4*)(inout + (long long)gm * F_DIM))[c4];
        float* dst = &As[row * AS_STRIDE + c4 * 4];
        dst[0] = v.x * inv;
        dst[1] = v.y * inv;
        dst[2] = v.z * inv;
        dst[3] = v.w * inv;
    }
    __syncthreads();

    const int lane = tid & 31;
    const int wv   = tid >> 5;     // wave 0..7 -> output column tile
    const int c0   = wv * 16;
    const int half = lane >> 4;    // 0: K+0/K+1 lanes, 1: K+2/K+3 lanes
    const int r    = lane & 15;

    const float bv = bias[c0 + r];

    v8f acc = {};
#pragma unroll
    for (int k = 0; k < F_DIM; k += 4) {
        // A 16x4 f32 fragment: lanes 0-15 -> (M=r, K=k,k+1); 16-31 -> (M=r, K=k+2,k+3)
        v2f a;
        a.x = As[r * AS_STRIDE + k + 2 * half + 0];
        a.y = As[r * AS_STRIDE + k + 2 * half + 1];
        // B 4x16 f32 fragment: v0 = rows K=k / K=k+2, v1 = rows K=k+1 / K=k+3
        const float* wp = weight + (long long)(k + 2 * half) * F_DIM + c0 + r;
        v2f b;
        b.x = wp[0];
        b.y = wp[F_DIM];
        acc = __builtin_amdgcn_wmma_f32_16x16x4_f32(
            /*neg_a=*/false, a, /*neg_b=*/false, b,
            /*c_mod=*/(short)0, acc, /*reuse_a=*/false, /*reuse_b=*/false);
    }

    // C/D layout: VGPR i -> M = i + 8*half, N = c0 + r. Fuse bias + ReLU.
#pragma unroll
    for (int i = 0; i < 8; ++i) {
        const int row = m0 + i + 8 * half;
        float v = acc[i] + bv;
        inout[(long long)row * F_DIM + c0 + r] = v > 0.0f ? v : 0.0f;
    }
}

// ---------------------------------------------------------------------------
extern "C" void kernel_launch(void* const* d_in, const int* in_sizes, int n_in,
                              void* d_out, int out_size, void* d_ws, size_t ws_size,
                              hipStream_t stream) {
    const float*     nodes  = (const float*)d_in[0];
    const long long* edges  = (const long long*)d_in[1];
    const float*     weight = (const float*)d_in[2];
    const float*     bias   = (const float*)d_in[3];

    float* out    = (float*)d_out;   // doubles as the aggregation buffer
    float* degree = (float*)d_ws;    // N floats of workspace

    const int n_edges      = in_sizes[1] / 2;
    const long long total  = (long long)BATCH * N_NODES * F_DIM;

    gcn_init_kernel<<<2048, 256, 0, stream>>>(out, degree, total, N_NODES);

    const int scatter_blocks = (n_edges + 7) / 8;   // 8 waves (edges) per block
    gcn_scatter_kernel<<<scatter_blocks, 256, 0, stream>>>(nodes, edges, out,
                                                           degree, n_edges);

    const int gemm_blocks = (BATCH * N_NODES) / 16; // 6250 tiles of 16 rows
    gcn_gemm_kernel<<<gemm_blocks, 256, 0, stream>>>(out, weight, bias, degree);
}